// VariateGeoHCAN_90958817395317
// MI455X (gfx1250) — compile-verified
//
#include <hip/hip_runtime.h>
#include <hip/hip_bf16.h>
#include <math.h>

// Problem constants (match reference)
#define D_MODEL   512
#define D_PROJ    32
#define N_CENT    6
#define CORR_H    32
#define CORR_IN   (D_MODEL + D_PROJ)   // 544
#define BATCH     64
#define NVAR      1024
#define NTOK      (BATCH * NVAR)       // 65536

typedef float v2f __attribute__((ext_vector_type(2)));
typedef float v8f __attribute__((ext_vector_type(8)));

#define V8_ZERO {0.f,0.f,0.f,0.f,0.f,0.f,0.f,0.f}

static __device__ __forceinline__ v8f wmma_f32_k4(v2f a, v2f b, v8f c) {
  // D = A(16x4,f32) * B(4x16,f32) + C(16x16,f32); wave32, fp32 WMMA
  return __builtin_amdgcn_wmma_f32_16x16x4_f32(false, a, false, b, (short)0, c,
                                               false, false);
}

static __device__ __forceinline__ float softplusf(float x) {
  return (x > 20.f) ? x : log1pf(expf(x));
}
static __device__ __forceinline__ float gelu_exact(float x) {
  return 0.5f * x * (1.f + erff(x * 0.70710678118654752440f));
}

// ---------------------------------------------------------------------------
// Kernel 0: zero the soft-assign accumulator in workspace (graph-safe init)
// ---------------------------------------------------------------------------
__global__ void zero_sa_kernel(float* __restrict__ sa) {
  if (threadIdx.x < N_CENT) sa[threadIdx.x] = 0.f;
}

// ---------------------------------------------------------------------------
// Kernel 1: RevIN stats. One thread per (b,d): mean + unbiased std over N.
// grid = B*D/256 = 128 blocks of 256.  Coalesced: consecutive tid -> d.
// ---------------------------------------------------------------------------
__global__ __launch_bounds__(256)
void revin_stats_kernel(const float* __restrict__ x,
                        float* __restrict__ mean_o,
                        float* __restrict__ std_o) {
  const int idx = blockIdx.x * 256 + threadIdx.x;       // b*512 + d
  const int b = idx >> 9;
  const int d = idx & (D_MODEL - 1);
  const float* p = x + (size_t)b * NVAR * D_MODEL + d;
  float s = 0.f, ss = 0.f;
  for (int n = 0; n < NVAR; ++n) {
    float v = p[(size_t)n * D_MODEL];
    s += v; ss += v * v;
  }
  const float mu  = s * (1.f / NVAR);
  float var = (ss - (float)NVAR * mu * mu) * (1.f / (NVAR - 1));
  mean_o[idx] = mu;
  std_o[idx]  = sqrtf(fmaxf(var, 0.f)) + 1e-5f;
}

// ---------------------------------------------------------------------------
// LDS layout (floats) for the fused kernel: 32 tokens per block, 2 waves.
// ---------------------------------------------------------------------------
#define OFF_XN     0                   // 32*512 = 16384
#define OFF_Q      16384               // 32*32  = 1024
#define OFF_CTX    17408               // 1024
#define OFF_H      18432               // 1024
#define OFF_MEAN   19456               // 512
#define OFF_STD    19968               // 512
#define OFF_CENT   20480               // 6*32 = 192
#define OFF_CNORM  20672               // 8
#define OFF_PROJB  20680               // 32
#define OFF_LNG    20712               // 32
#define OFF_LNB    20744               // 32
#define OFF_SA     20776               // 8
#define SMEM_FLOATS 20784
#define SMEM_BYTES  (SMEM_FLOATS * 4)  // 83136 B dynamic LDS

// ---------------------------------------------------------------------------
// Kernel 2: fully fused per-token pipeline. 64 threads (2 waves), each wave
// owns 16 tokens; three fp32 WMMA GEMMs + per-token LN/softmax section.
// grid = NTOK/32 = 2048 blocks.
// ---------------------------------------------------------------------------
__global__ __launch_bounds__(64)
void fused_token_kernel(const float* __restrict__ x,
                        const float* __restrict__ proj_w,     // [32,512]
                        const float* __restrict__ proj_b,     // [32]
                        const float* __restrict__ ln_g,       // [32]
                        const float* __restrict__ ln_b,       // [32]
                        const float* __restrict__ cent,       // [6,32]
                        const float* __restrict__ ciw,        // [32,544]
                        const float* __restrict__ cib,        // [32]
                        const float* __restrict__ cow,        // [512,32]
                        const float* __restrict__ cob,        // [512]
                        const float* __restrict__ gate_alpha, // [1]
                        const float* __restrict__ log_temp,   // [1]
                        const float* __restrict__ scale_ev,   // [1]
                        const float* __restrict__ rev_mean,   // [B*512]
                        const float* __restrict__ rev_std,    // [B*512]
                        float* __restrict__ sa_global,        // [6] accumulator
                        float* __restrict__ out,              // [B*N*512]
                        float* __restrict__ u_std)            // [B*N]
{
  extern __shared__ float lds[];
  const int tid  = threadIdx.x;
  const int w    = tid >> 5;          // wave id (0..1)
  const int lane = tid & 31;
  const int lo   = lane & 15;
  const int hi   = lane >> 4;
  const int tokbase = blockIdx.x * 32;       // 32 consecutive tokens (same b)
  const int b = tokbase >> 10;               // 1024 tokens per batch entry

  // ---- Phase 0: preload broadcast data into LDS --------------------------
  for (int i = tid; i < D_MODEL; i += 64) {
    lds[OFF_MEAN + i] = rev_mean[b * D_MODEL + i];
    lds[OFF_STD  + i] = rev_std [b * D_MODEL + i];
  }
  for (int i = tid; i < N_CENT * D_PROJ; i += 64) lds[OFF_CENT + i] = cent[i];
  if (tid < D_PROJ) {
    lds[OFF_PROJB + tid] = proj_b[tid];
    lds[OFF_LNG   + tid] = ln_g[tid];
    lds[OFF_LNB   + tid] = ln_b[tid];
  }
  if (tid < N_CENT) lds[OFF_SA + tid] = 0.f;
  __syncthreads();

  // ---- Phase 1: load x tile, normalize into LDS; centroid norms ----------
  // x tile is [32 tokens][512]; idx walks it row-major => coalesced HBM reads
  {
    const size_t xbase = (size_t)tokbase * D_MODEL;
    for (int i = tid; i < 32 * D_MODEL; i += 64) {
      const int d = i & (D_MODEL - 1);
      lds[OFF_XN + i] = (x[xbase + i] - lds[OFF_MEAN + d]) / lds[OFF_STD + d];
    }
  }
  if (tid < N_CENT) {
    float s = 0.f;
    for (int p = 0; p < D_PROJ; ++p) {
      float c = lds[OFF_CENT + tid * D_PROJ + p];
      s += c * c;
    }
    lds[OFF_CNORM + tid] = fmaxf(sqrtf(s), 1e-12f);
  }
  __syncthreads();

  // ---- Phase 2: GEMM1  Q[16x32] = x_norm(16x512) @ proj_w^T  (WMMA fp32) --
  {
    v8f acc0 = V8_ZERO, acc1 = V8_ZERO;
    const float* arow = lds + OFF_XN + (w * 16 + lo) * D_MODEL + 2 * hi;
    const float* b0p  = proj_w + (size_t)lo        * D_MODEL + 2 * hi;
    const float* b1p  = proj_w + (size_t)(16 + lo) * D_MODEL + 2 * hi;
    for (int k = 0; k < D_MODEL; k += 4) {
      v2f a  = *(const v2f*)(arow + k);
      v2f b0 = *(const v2f*)(b0p + k);
      v2f b1 = *(const v2f*)(b1p + k);
      acc0 = wmma_f32_k4(a, b0, acc0);
      acc1 = wmma_f32_k4(a, b1, acc1);
    }
#pragma unroll
    for (int r = 0; r < 8; ++r) {
      const int m = w * 16 + r + 8 * hi;
      lds[OFF_Q + m * D_PROJ + lo]      = acc0[r];
      lds[OFF_Q + m * D_PROJ + 16 + lo] = acc1[r];
    }
  }
  __syncthreads();

  // ---- Phase 3: per-token LN -> cosine routing -> softmax -> ctx, u_std --
  if (hi == 0) {                       // 16 lanes/wave, one token each
    const int tk  = w * 16 + lo;       // local token row
    const int tok = tokbase + tk;
    float q[D_PROJ];
    float mu = 0.f;
#pragma unroll
    for (int p = 0; p < D_PROJ; ++p) {
      q[p] = lds[OFF_Q + tk * D_PROJ + p] + lds[OFF_PROJB + p];
      mu += q[p];
    }
    mu *= (1.f / D_PROJ);
    float var = 0.f;
#pragma unroll
    for (int p = 0; p < D_PROJ; ++p) { float d = q[p] - mu; var += d * d; }
    var *= (1.f / D_PROJ);
    const float rs = rsqrtf(var + 1e-5f);
    float n2 = 0.f;
#pragma unroll
    for (int p = 0; p < D_PROJ; ++p) {
      q[p] = (q[p] - mu) * rs * lds[OFF_LNG + p] + lds[OFF_LNB + p];
      n2 += q[p] * q[p];
    }
    const float inv_qn = 1.f / fmaxf(sqrtf(n2), 1e-12f);

    float sim[N_CENT];
    float sim_mean = 0.f;
#pragma unroll
    for (int k = 0; k < N_CENT; ++k) {
      float dot = 0.f;
#pragma unroll
      for (int p = 0; p < D_PROJ; ++p) dot += q[p] * lds[OFF_CENT + k * D_PROJ + p];
      sim[k] = dot * inv_qn / lds[OFF_CNORM + k];
      sim_mean += sim[k];
    }
    sim_mean *= (1.f / N_CENT);

    // softmax(sim * exp(log_temp))
    const float temp = expf(log_temp[0]);
    float mx = -1e30f;
#pragma unroll
    for (int k = 0; k < N_CENT; ++k) mx = fmaxf(mx, sim[k] * temp);
    float sa[N_CENT]; float se = 0.f;
#pragma unroll
    for (int k = 0; k < N_CENT; ++k) { sa[k] = expf(sim[k] * temp - mx); se += sa[k]; }
    const float ise = 1.f / se;
#pragma unroll
    for (int k = 0; k < N_CENT; ++k) sa[k] *= ise;

    // centroid context into LDS
#pragma unroll
    for (int p = 0; p < D_PROJ; ++p) {
      float c = 0.f;
#pragma unroll
      for (int k = 0; k < N_CENT; ++k) c += sa[k] * lds[OFF_CENT + k * D_PROJ + p];
      lds[OFF_CTX + tk * D_PROJ + p] = c;
    }

    // DST uncertainty diagnostic
    const float s_ev = softplusf(scale_ev[0]);
    float sdir = 0.f;
#pragma unroll
    for (int k = 0; k < N_CENT; ++k) sdir += softplusf((sim[k] - sim_mean) * 5.0f);
    u_std[tok] = (float)N_CENT / ((float)N_CENT + s_ev * sdir);

    // routing-mean accumulator (block-local)
#pragma unroll
    for (int k = 0; k < N_CENT; ++k) atomicAdd(&lds[OFF_SA + k], sa[k]);
  }
  __syncthreads();

  // ---- Phase 4: GEMM2  h = gelu(concat(x_norm,ctx)(16x544) @ ciw^T) ------
  {
    v8f acc0 = V8_ZERO, acc1 = V8_ZERO;
    const float* xrow = lds + OFF_XN  + (w * 16 + lo) * D_MODEL + 2 * hi;
    const float* crow = lds + OFF_CTX + (w * 16 + lo) * D_PROJ  + 2 * hi;
    const float* b0p  = ciw + (size_t)lo        * CORR_IN + 2 * hi;
    const float* b1p  = ciw + (size_t)(16 + lo) * CORR_IN + 2 * hi;
    for (int k = 0; k < D_MODEL; k += 4) {
      v2f a  = *(const v2f*)(xrow + k);
      v2f b0 = *(const v2f*)(b0p + k);
      v2f b1 = *(const v2f*)(b1p + k);
      acc0 = wmma_f32_k4(a, b0, acc0);
      acc1 = wmma_f32_k4(a, b1, acc1);
    }
    for (int k = D_MODEL; k < CORR_IN; k += 4) {
      v2f a  = *(const v2f*)(crow + (k - D_MODEL));
      v2f b0 = *(const v2f*)(b0p + k);
      v2f b1 = *(const v2f*)(b1p + k);
      acc0 = wmma_f32_k4(a, b0, acc0);
      acc1 = wmma_f32_k4(a, b1, acc1);
    }
    const float bi0 = cib[lo], bi1 = cib[16 + lo];
#pragma unroll
    for (int r = 0; r < 8; ++r) {
      const int m = w * 16 + r + 8 * hi;
      lds[OFF_H + m * CORR_H + lo]      = gelu_exact(acc0[r] + bi0);
      lds[OFF_H + m * CORR_H + 16 + lo] = gelu_exact(acc1[r] + bi1);
    }
  }
  __syncthreads();

  // ---- Phase 5: GEMM3 correction(16x512) = h(16x32) @ cow^T + epilogue ---
  {
    const float gate = 1.f / (1.f + expf(-gate_alpha[0]));
    const float* hrow = lds + OFF_H + (w * 16 + lo) * CORR_H + 2 * hi;
    for (int nt = 0; nt < D_MODEL / 16; ++nt) {
      const int n0 = nt * 16 + lo;                      // output feature d
      const float* bp = cow + (size_t)n0 * CORR_H + 2 * hi;
      v8f acc = V8_ZERO;
#pragma unroll
      for (int k = 0; k < CORR_H; k += 4) {
        acc = wmma_f32_k4(*(const v2f*)(hrow + k), *(const v2f*)(bp + k), acc);
      }
      const float bo = cob[n0];
      const float mn = lds[OFF_MEAN + n0];
      const float sd = lds[OFF_STD + n0];
#pragma unroll
      for (int r = 0; r < 8; ++r) {
        const int m  = r + 8 * hi;
        const int tk = w * 16 + m;
        const float xnv = lds[OFF_XN + tk * D_MODEL + n0];
        // out = x + gate*corr  with  x = xn*std+mean, corr = (raw+b)*std
        out[(size_t)(tokbase + tk) * D_MODEL + n0] =
            mn + sd * (xnv + gate * (acc[r] + bo));
      }
    }
  }
  __syncthreads();

  if (tid < N_CENT) atomicAdd(&sa_global[tid], lds[OFF_SA + tid]);
}

// ---------------------------------------------------------------------------
// Kernel 3: scalar losses (tiny).
// ---------------------------------------------------------------------------
__global__ void loss_kernel(const float* __restrict__ cent,
                            const float* __restrict__ sa_sum,
                            float* __restrict__ ortho,
                            float* __restrict__ latent) {
  if (blockIdx.x == 0 && threadIdx.x == 0) {
    float cn[N_CENT][D_PROJ];
    for (int i = 0; i < N_CENT; ++i) {
      float s = 0.f;
      for (int p = 0; p < D_PROJ; ++p) { float v = cent[i * D_PROJ + p]; s += v * v; }
      float inv = 1.f / fmaxf(sqrtf(s), 1e-12f);
      for (int p = 0; p < D_PROJ; ++p) cn[i][p] = cent[i * D_PROJ + p] * inv;
    }
    float acc = 0.f;
    for (int i = 0; i < N_CENT; ++i)
      for (int j = 0; j < N_CENT; ++j) {
        float g = 0.f;
        for (int p = 0; p < D_PROJ; ++p) g += cn[i][p] * cn[j][p];
        float t = g - ((i == j) ? 1.f : 0.f);
        acc += t * t;
      }
    *ortho = acc * (1.f / (N_CENT * N_CENT));

    float l = 0.f;
    for (int k = 0; k < N_CENT; ++k) {
      float av = sa_sum[k] * (1.f / (float)NTOK) - (1.f / (float)N_CENT);
      l += av * av;
    }
    *latent = l * (1.f / (float)N_CENT);
  }
}

// ---------------------------------------------------------------------------
extern "C" void kernel_launch(void* const* d_in, const int* in_sizes, int n_in,
                              void* d_out, int out_size, void* d_ws, size_t ws_size,
                              hipStream_t stream) {
  const float* x          = (const float*)d_in[0];
  const float* proj_w     = (const float*)d_in[1];
  const float* proj_b     = (const float*)d_in[2];
  const float* ln_g       = (const float*)d_in[3];
  const float* ln_b       = (const float*)d_in[4];
  const float* cent       = (const float*)d_in[5];
  const float* ciw        = (const float*)d_in[6];
  const float* cib        = (const float*)d_in[7];
  const float* cow        = (const float*)d_in[8];
  const float* cob        = (const float*)d_in[9];
  const float* gate_alpha = (const float*)d_in[10];
  const float* log_temp   = (const float*)d_in[11];
  const float* scale_ev   = (const float*)d_in[12];

  float* ws      = (float*)d_ws;
  float* ws_mean = ws;                        // B*512
  float* ws_std  = ws + BATCH * D_MODEL;      // B*512
  float* ws_sa   = ws + 2 * BATCH * D_MODEL;  // 6

  float* out    = (float*)d_out;              // [B,N,512]
  float* u_std  = out + (size_t)NTOK * D_MODEL;
  float* ortho  = u_std + NTOK;
  float* latent = ortho + 1;

  zero_sa_kernel<<<1, 32, 0, stream>>>(ws_sa);
  revin_stats_kernel<<<(BATCH * D_MODEL) / 256, 256, 0, stream>>>(x, ws_mean, ws_std);
  fused_token_kernel<<<NTOK / 32, 64, SMEM_BYTES, stream>>>(
      x, proj_w, proj_b, ln_g, ln_b, cent, ciw, cib, cow, cob,
      gate_alpha, log_temp, scale_ev, ws_mean, ws_std, ws_sa, out, u_std);
  loss_kernel<<<1, 32, 0, stream>>>(cent, ws_sa, ortho, latent);
}